// BCE_for_non_zero_81277961109765
// MI455X (gfx1250) — compile-verified
//
#include <hip/hip_runtime.h>
#include <hip/hip_bf16.h>

typedef float v2f __attribute__((ext_vector_type(2)));
typedef float v8f __attribute__((ext_vector_type(8)));

#define BLOCKS      2048
#define THREADS     256
#define STRIDE      (BLOCKS * THREADS)   // 524288 = 2^19 lanes (compile-time)

// Numerically stable BCE-with-logits element:
//   log(1+exp(x)) - x*t  ==  max(x,0) - x*t + log1p(exp(-|x|))
// Hardware v_exp_f32 / v_log_f32 (quarter-rate TRANS, below the HBM floor).
__device__ __forceinline__ float bce_elem(float x, float t) {
    float e = __expf(-fabsf(x));           // in (0, 1]
    return fmaxf(x, 0.0f) - x * t + __logf(1.0f + e);
}

// Wave32 full reduction of (acc0 + acc1) across all 32 lanes using one
// V_WMMA_F32_16X16X4_F32 with B = ones(4x16):
//   A (16x4 f32): lane m   holds A[m][0]=vgpr0, A[m][1]=vgpr1 (m=0..15)
//                 lane m+16 holds A[m][2]=vgpr0, A[m][3]=vgpr1
//   D[m][n] = sum_k A[m][k]; D vgpr j: lanes 0-15 = row j, lanes 16-31 = row j+8.
// Sum of the 8 D vgprs gives rows 0-7 (lanes<16) / rows 8-15 (lanes>=16);
// one shfl_xor(16) finishes the 64-partial reduction. EXEC must be all-1s here.
__device__ __forceinline__ float wave_reduce_wmma(float acc0, float acc1) {
    v2f a; a[0] = acc0; a[1] = acc1;
    v2f b; b[0] = 1.0f; b[1] = 1.0f;
    v8f d = {};
    d = __builtin_amdgcn_wmma_f32_16x16x4_f32(
        /*neg_a=*/false, a, /*neg_b=*/false, b,
        /*c_mod=*/(short)0, d, /*reuse_a=*/false, /*reuse_b=*/false);
    float r = d[0] + d[1] + d[2] + d[3] + d[4] + d[5] + d[6] + d[7];
    r += __shfl_xor(r, 16, 32);
    return r;
}

__global__ __launch_bounds__(THREADS) void bce_partial_kernel(
    const float4* __restrict__ in, const float4* __restrict__ tg,
    float* __restrict__ partials, int n4)
{
    const int tid = blockIdx.x * THREADS + threadIdx.x;

    // Uniform trip count (SALU-only loop control): STRIDE is a compile-time
    // power of two, so iters/rem are simple shifts/masks, identical per lane.
    const int iters = n4 >> 19;            // n4 / STRIDE
    const int rem   = n4 & (STRIDE - 1);   // n4 % STRIDE

    // Two independent accumulator chains per lane (ILP) -> WMMA A vgpr0/vgpr1.
    float acc0 = 0.0f, acc1 = 0.0f;

    int i = tid;
    #pragma unroll 4
    for (int j = 0; j < iters; ++j, i += STRIDE) {
        float4 x = in[i];   // global_load_b128: rows 2i (x.x,x.y), 2i+1 (x.z,x.w)
        float4 t = tg[i];   // global_load_b128
        float m0 = (t.x + t.y > 0.0f) ? 1.0f : 0.0f;   // row-any-positive mask
        float m1 = (t.z + t.w > 0.0f) ? 1.0f : 0.0f;
        acc0 += m0 * (bce_elem(x.x, t.x) + bce_elem(x.y, t.y));
        acc1 += m1 * (bce_elem(x.z, t.z) + bce_elem(x.w, t.w));
    }
    if (tid < rem) {        // single divergent tail iteration (if any)
        const int k = iters * STRIDE + tid;
        float4 x = in[k];
        float4 t = tg[k];
        float m0 = (t.x + t.y > 0.0f) ? 1.0f : 0.0f;
        float m1 = (t.z + t.w > 0.0f) ? 1.0f : 0.0f;
        acc0 += m0 * (bce_elem(x.x, t.x) + bce_elem(x.y, t.y));
        acc1 += m1 * (bce_elem(x.z, t.z) + bce_elem(x.w, t.w));
    }

    // EXEC is all-1s again here (tail 'if' is structurally closed).
    float r = wave_reduce_wmma(acc0, acc1);   // full wave32 sum in every lane

    __shared__ float wsum[THREADS / 32];      // 8 wave32s per block
    const int lane = threadIdx.x & 31;
    const int wave = threadIdx.x >> 5;
    if (lane == 0) wsum[wave] = r;
    __syncthreads();
    if (threadIdx.x == 0) {
        float s = 0.0f;
        #pragma unroll
        for (int w = 0; w < THREADS / 32; ++w) s += wsum[w];
        partials[blockIdx.x] = s;
    }
}

__global__ __launch_bounds__(THREADS) void bce_final_kernel(
    const float* __restrict__ partials, int np,
    float* __restrict__ out, float inv_n)
{
    float s = 0.0f;
    for (int i = threadIdx.x; i < np; i += THREADS) s += partials[i];
    #pragma unroll
    for (int off = 16; off > 0; off >>= 1) s += __shfl_xor(s, off, 32);
    __shared__ float wsum[THREADS / 32];
    const int lane = threadIdx.x & 31;
    const int wave = threadIdx.x >> 5;
    if (lane == 0) wsum[wave] = s;
    __syncthreads();
    if (threadIdx.x == 0) {
        float tot = 0.0f;
        #pragma unroll
        for (int w = 0; w < THREADS / 32; ++w) tot += wsum[w];
        out[0] = tot * inv_n;   // mean over all B*C elements (ALPHA == 1.0)
    }
}

extern "C" void kernel_launch(void* const* d_in, const int* in_sizes, int n_in,
                              void* d_out, int out_size, void* d_ws, size_t ws_size,
                              hipStream_t stream) {
    const float4* in = (const float4*)d_in[0];   // inputs  [B, 2] f32
    const float4* tg = (const float4*)d_in[1];   // targets [B, 2] f32
    const int n  = in_sizes[0];                  // B*C = 16,777,216 elements
    const int n4 = n >> 2;                       // float4 count (two rows each)

    float* partials = (float*)d_ws;              // BLOCKS floats of scratch

    bce_partial_kernel<<<BLOCKS, THREADS, 0, stream>>>(in, tg, partials, n4);
    bce_final_kernel<<<1, THREADS, 0, stream>>>(partials, BLOCKS, (float*)d_out,
                                                1.0f / (float)n);
}